// VC_model_79156247266099
// MI455X (gfx1250) — compile-verified
//
#include <hip/hip_runtime.h>

// Problem constants (from reference): inputs (B=8, S=512, A=96, V=2) f32,
// output (B, S-1=511, A, A) f32. WIN=10 -> K = WIN*V = 20.
#define B_DIM   8
#define S_DIM   512
#define A_DIM   96
#define V_DIM   2
#define WINSZ   10
#define KTOT    (WINSZ * V_DIM)   // 20
#define TOUT    (S_DIM - 1)       // 511
#define JW      (A_DIM * V_DIM)   // 192 = (atom,feature) pairs per s-row
#define LDSTR   104               // LDS row stride (floats); 2*104 % 64 = 16 -> no bank conflict between lane halves

typedef __attribute__((ext_vector_type(2))) float v2f;
typedef __attribute__((ext_vector_type(8))) float v8f;

// ---------------------------------------------------------------------------
// Kernel 1: per-(b, atom, feature) mean over the S axis. 8 blocks x 192 thr.
// ---------------------------------------------------------------------------
__global__ __launch_bounds__(JW) void vc_mean_kernel(const float* __restrict__ x,
                                                     float* __restrict__ mean) {
    const int b = blockIdx.x;
    const int j = threadIdx.x;                 // 0..191, coalesced over (i,v)
    const float* xb = x + (size_t)b * S_DIM * JW;
    float s = 0.0f;
    #pragma unroll 4
    for (int ss = 0; ss < S_DIM; ++ss)
        s += xb[(size_t)ss * JW + j];
    mean[b * JW + j] = s * (1.0f / (float)S_DIM);
}

// ---------------------------------------------------------------------------
// Kernel 2: one workgroup per (b, t). 192 threads = 6 waves.
//   LDS Xw[k][i]: mean-centered window, k = 2*s_off + v, zero-padded to K=20.
//   W = Xw^T Xw per 16x16 tile via V_WMMA_F32_16X16X4_F32 (5 chained, K=20).
//   Wave w owns row block m0 = w*16 (A-fragments + diag rsqrts loop-invariant)
//   and sweeps the 6 column blocks -> writes a contiguous 16x96 output stripe.
//   out[i][j] = W[i][j] * rsqrt(W[i][i]) * rsqrt(W[j][j]), diag masked to 0.
// ---------------------------------------------------------------------------
__global__ __launch_bounds__(JW) void vc_windowed_corr_kernel(const float* __restrict__ x,
                                                              const float* __restrict__ mean,
                                                              float* __restrict__ out) {
    __shared__ float Xw[KTOT][LDSTR];
    __shared__ float rd[A_DIM];

    const int t = blockIdx.x;       // 0..510
    const int b = blockIdx.y;       // 0..7
    const int j = threadIdx.x;      // 0..191

    // Zero-pad LDS window (rows beyond the valid window must contribute 0).
    for (int p = j; p < KTOT * LDSTR; p += JW)
        (&Xw[0][0])[p] = 0.0f;
    __syncthreads();

    // Fill valid window rows: window for output t is s in [max(0, t-9), t].
    const int i = j >> 1;
    const int v = j & 1;
    const float mu = mean[b * JW + j];
    const float* xb = x + (size_t)b * S_DIM * JW;
    for (int r = 0; r < WINSZ; ++r) {
        const int s = t - r;
        if (s < 0) break;
        Xw[2 * r + v][i] = xb[(size_t)s * JW + j] - mu;
    }
    __syncthreads();

    // Diagonal: d[i] = sum_k Xw[k][i]^2 ; store rsqrt(d).
    if (j < A_DIM) {
        float d = 0.0f;
        #pragma unroll
        for (int k = 0; k < KTOT; ++k) {
            const float w = Xw[k][j];
            d += w * w;
        }
        rd[j] = __frsqrt_rn(d);
    }
    __syncthreads();

    const int lane = j & 31;
    const int wave = j >> 5;        // 0..5
    const int lm   = lane & 15;     // lane-in-half
    const int half = lane >> 4;     // 0: K pair {0,1}; 1: K pair {2,3}

    const int m0 = wave * 16;       // row block owned by this wave (invariant)

    // Hoist A fragments for this row block: F32 16x4 A layout puts
    // VGPR0 = A[lm, kb], VGPR1 = A[lm, kb+1] with kb = 4*kc + 2*half.
    v2f afrag[KTOT / 4];
    #pragma unroll
    for (int kc = 0; kc < KTOT / 4; ++kc) {
        const int kb = 4 * kc + 2 * half;
        afrag[kc][0] = Xw[kb][m0 + lm];
        afrag[kc][1] = Xw[kb + 1][m0 + lm];
    }

    // Hoist the 8 diagonal rsqrts this half-lane needs: D-matrix VGPR r maps
    // to row m = m0 + 8*half + r, i.e. 8 consecutive floats -> two b128 loads.
    float rdm[8];
    {
        const float4 rA = *(const float4*)&rd[m0 + 8 * half];
        const float4 rB = *(const float4*)&rd[m0 + 8 * half + 4];
        rdm[0] = rA.x; rdm[1] = rA.y; rdm[2] = rA.z; rdm[3] = rA.w;
        rdm[4] = rB.x; rdm[5] = rB.y; rdm[6] = rB.z; rdm[7] = rB.w;
    }

    const size_t obase = ((size_t)b * TOUT + t) * (size_t)(A_DIM * A_DIM);

    #pragma unroll
    for (int q = 0; q < A_DIM / 16; ++q) {      // sweep 6 column blocks
        const int n0 = q * 16;

        v8f c = {};
        #pragma unroll
        for (int kc = 0; kc < KTOT / 4; ++kc) {
            // B (4x16) layout mirrors A's for X * X^T.
            const int kb = 4 * kc + 2 * half;
            v2f bm;
            bm[0] = Xw[kb][n0 + lm];
            bm[1] = Xw[kb + 1][n0 + lm];
            c = __builtin_amdgcn_wmma_f32_16x16x4_f32(
                    /*neg_a=*/false, afrag[kc], /*neg_b=*/false, bm,
                    /*c_mod=*/(short)0, c, /*reuse_a=*/false, /*reuse_b=*/false);
        }

        // Branchless epilogue: normalize, mask diagonal via register select,
        // store. 16 lanes per half write one 64B run of an output row.
        const int   n   = n0 + lm;
        const float rdn = rd[n];
        #pragma unroll
        for (int r = 0; r < 8; ++r) {
            const int m   = m0 + 8 * half + r;
            float     val = c[r] * rdm[r] * rdn;
            val = (m != n) ? val : 0.0f;
            out[obase + (size_t)m * A_DIM + n] = val;
        }
    }
}

// ---------------------------------------------------------------------------
extern "C" void kernel_launch(void* const* d_in, const int* in_sizes, int n_in,
                              void* d_out, int out_size, void* d_ws, size_t ws_size,
                              hipStream_t stream) {
    (void)in_sizes; (void)n_in; (void)out_size; (void)ws_size;
    const float* x   = (const float*)d_in[0];
    float*       out = (float*)d_out;
    float*       mu  = (float*)d_ws;          // B_DIM * JW floats = 6 KB

    vc_mean_kernel<<<B_DIM, JW, 0, stream>>>(x, mu);
    vc_windowed_corr_kernel<<<dim3(TOUT, B_DIM), JW, 0, stream>>>(x, mu, out);
}